// QuaternionMaxPool2D_47931835023631
// MI455X (gfx1250) — compile-verified
//
#include <hip/hip_runtime.h>

typedef __attribute__((ext_vector_type(4))) float f4;
typedef __attribute__((ext_vector_type(2))) float f2;

// Compile-time shape (reference: x[16,128,256,256] f32, pool 2x2 stride 2).
constexpr int B  = 16;
constexpr int C  = 128;
constexpr int Q  = C / 4;      // 32
constexpr int H  = 256;
constexpr int W  = 256;
constexpr int HN = H / 2;      // 128
constexpr int WN = W / 2;      // 128
constexpr int WN2 = WN / 2;    // 64 : output-column pairs per row
constexpr int HO_BITS = 7;     // log2(HN)
constexpr int Q_BITS  = 5;     // log2(Q)

// One 64-thread block (2 wave32 waves) handles one (b, q, ho) output row.
// All base addresses derive from blockIdx.x only -> SALU; the only per-lane
// address term is threadIdx.x*16B (loads) / *8B (stores), so loads/stores use SADDR + voffset form.
// Each thread: 2 output columns; reads 2 rows x 4 planes as b128 NT (128 B),
// writes 4 component f2 pairs as b64 NT (32 B).
__global__ __launch_bounds__(64) void qmaxpool2x2_kernel(
    const float* __restrict__ x, float* __restrict__ out) {
  // Scalar (uniform) decomposition of the block index.
  const int ho = blockIdx.x & (HN - 1);
  const int bq = blockIdx.x >> HO_BITS;          // b*Q + q
  const int q  = bq & (Q - 1);
  const int b  = bq >> Q_BITS;
  const int chan0 = b * 4 * Q + q;               // component-0 channel index

  constexpr int planeStride = Q * H * W;         // elems between components
  const float* __restrict__ rowp =
      x + (size_t)chan0 * (H * W) + (2 * ho) * W;  // uniform base (SGPRs)

  const int wo2 = threadIdx.x;                   // 0..63, per-lane
  const int off = 4 * wo2;                       // elems; 16B-aligned lane offset

  // Load 2x2 windows for both output columns, all 4 components.
  f4 r0[4], r1[4];
#pragma unroll
  for (int c = 0; c < 4; ++c) {
    const float* p = rowp + c * planeStride + off;
    r0[c] = __builtin_nontemporal_load((const f4*)p);        // row h0
    r1[c] = __builtin_nontemporal_load((const f4*)(p + W));  // row h0+1
  }

  float o0[4], o1[4];
#pragma unroll
  for (int p = 0; p < 2; ++p) {
    // Window values in torch-unfold order: (0,0),(0,1),(1,0),(1,1)
    float v[4][4];
#pragma unroll
    for (int c = 0; c < 4; ++c) {
      v[c][0] = r0[c][2 * p + 0];
      v[c][1] = r0[c][2 * p + 1];
      v[c][2] = r1[c][2 * p + 0];
      v[c][3] = r1[c][2 * p + 1];
    }
    // Squared amplitude per window position (sqrt is monotonic -> skip it).
    float amp[4];
#pragma unroll
    for (int k = 0; k < 4; ++k)
      amp[k] = v[0][k] * v[0][k] + v[1][k] * v[1][k] +
               v[2][k] * v[2][k] + v[3][k] * v[3][k];

    // Tournament: carry running best amp^2 and best quaternion together.
    // Strict '>' keeps the first occurrence (matches jnp.argmax ties).
    float best = amp[0];
    float sel[4];
#pragma unroll
    for (int c = 0; c < 4; ++c) sel[c] = v[c][0];
#pragma unroll
    for (int k = 1; k < 4; ++k) {
      const bool g = amp[k] > best;
      best = g ? amp[k] : best;
#pragma unroll
      for (int c = 0; c < 4; ++c) sel[c] = g ? v[c][k] : sel[c];
    }
#pragma unroll
    for (int c = 0; c < 4; ++c) {
      if (p == 0) o0[c] = sel[c]; else o1[c] = sel[c];
    }
  }

  // Store: per component, the two adjacent output columns as one float2 (8B).
  constexpr int oplane = Q * HN * WN;            // 2 MiB/4 -> fits imm24 in bytes
  float* __restrict__ orow =
      out + (size_t)chan0 * (HN * WN) + ho * WN;   // uniform base (SGPRs)
#pragma unroll
  for (int c = 0; c < 4; ++c) {
    f2 o;
    o.x = o0[c];
    o.y = o1[c];
    __builtin_nontemporal_store(o, (f2*)(orow + c * oplane + 2 * wo2));
  }
}

extern "C" void kernel_launch(void* const* d_in, const int* in_sizes, int n_in,
                              void* d_out, int out_size, void* d_ws, size_t ws_size,
                              hipStream_t stream) {
  (void)in_sizes; (void)n_in; (void)d_ws; (void)ws_size; (void)out_size;

  const float* x = (const float*)d_in[0];
  float* out = (float*)d_out;

  constexpr int blocks  = B * Q * HN;  // 65,536 : one block per output row
  constexpr int threads = WN2;         // 64 = 2 wave32 waves

  qmaxpool2x2_kernel<<<blocks, threads, 0, stream>>>(x, out);
}